// UVNet_816043786540
// MI455X (gfx1250) — compile-verified
//
#include <hip/hip_runtime.h>

// Problem constants (fixed by the reference setup_inputs()).
#define B_    64
#define C_    256
#define H_    64
#define W_    64
#define HW_   4096            // H*W == uv_res*uv_res
#define NPIX  (B_ * HW_)      // 262144 pixels
#define NBINS (B_ * HW_)      // 262144 UV bins
#define OC_   (C_ + 3)        // 259 output channels
#define CH_PER_THREAD 32      // channel chunk per scatter thread

typedef float v4f __attribute__((ext_vector_type(4)));

__device__ __forceinline__ void atom_add_f32(float* p, float v) {
    // relaxed, agent scope -> plain global_atomic_add_f32 (no return, no CAS loop)
    __hip_atomic_fetch_add(p, v, __ATOMIC_RELAXED, __HIP_MEMORY_SCOPE_AGENT);
}

// ---------------------------------------------------------------------------
// Pass 1: scatter-add. grid = (NPIX/4/256, C/CH_PER_THREAD), block = 256.
// Thread = (4 consecutive pixels, 32-channel chunk). Consecutive threads ->
// consecutive 16B pixel groups, so conf/u/v and every feature-plane access is
// a coalesced global_load_b128. Feature data is single-use: NT hint so the
// 192MB L2 keeps capacity for the atomic accumulation region. Atomics are
// scalar by nature (4 distinct bins per thread).
// ---------------------------------------------------------------------------
__global__ __launch_bounds__(256) void uvnet_scatter(
    const float* __restrict__ dp,     // (B,3,H,W)
    const float* __restrict__ feat,   // (B,C,H,W)
    float* __restrict__ out,          // (B,259,64,64) accumulators (pre-zeroed)
    float* __restrict__ wsum,         // (B*4096)
    float* __restrict__ sxn,          // (B*4096)
    float* __restrict__ syn)          // (B*4096)
{
    const int p0  = (blockIdx.x * 256 + threadIdx.x) * 4;  // first pixel of group
    const int b   = p0 >> 12;
    const int pi0 = p0 & 4095;                             // y*64 + x, multiple of 4

    const float* dpb = dp + (size_t)b * (3 * HW_) + pi0;
    const v4f conf = *(const v4f*)(dpb);
    bool val0 = conf.x > 0.5f, val1 = conf.y > 0.5f,
         val2 = conf.z > 0.5f, val3 = conf.w > 0.5f;
    if (!(val0 | val1 | val2 | val3)) return;              // whole group invalid

    const v4f u4 = *(const v4f*)(dpb + HW_);
    const v4f v4 = *(const v4f*)(dpb + 2 * HW_);

    int bin[4];
    #pragma unroll
    for (int j = 0; j < 4; ++j) {
        int ur = (int)rintf(u4[j] * 63.0f); ur = ur < 0 ? 0 : (ur > 63 ? 63 : ur);
        int vr = (int)rintf(v4[j] * 63.0f); vr = vr < 0 ? 0 : (vr > 63 ? 63 : vr);
        bin[j] = (vr << 6) + ur;                           // v*64 + u
    }
    const bool valid[4] = {val0, val1, val2, val3};

    const int c0 = blockIdx.y * CH_PER_THREAD;
    if (blockIdx.y == 0) {
        // once-per-pixel scalar accumulations (weight, xn, yn)
        const int x0 = pi0 & 63, y0 = pi0 >> 6;            // x0 multiple of 4, same row
        const float yn = 2.0f * (float)y0 * (1.0f / 63.0f) - 1.0f;
        #pragma unroll
        for (int j = 0; j < 4; ++j) {
            if (valid[j]) {
                const float xn = 2.0f * (float)(x0 + j) * (1.0f / 63.0f) - 1.0f;
                const int g = (b << 12) + bin[j];
                atom_add_f32(wsum + g, 1.0f);
                atom_add_f32(sxn  + g, xn);
                atom_add_f32(syn  + g, yn);
            }
        }
    }

    const float* fptr = feat + (((size_t)(b * C_  + c0)) << 12) + pi0;
    float*       obase = out + (((size_t)(b * OC_ + c0)) << 12);
    #pragma unroll 4
    for (int c = 0; c < CH_PER_THREAD; ++c) {
        const v4f f = __builtin_nontemporal_load((const v4f*)fptr); // b128 NT stream
        if (val0) atom_add_f32(obase + bin[0], f.x);
        if (val1) atom_add_f32(obase + bin[1], f.y);
        if (val2) atom_add_f32(obase + bin[2], f.z);
        if (val3) atom_add_f32(obase + bin[3], f.w);
        fptr  += HW_;
        obase += HW_;
    }
}

// ---------------------------------------------------------------------------
// Pass 2: normalize in place. One thread per 4 consecutive UV bins; every
// per-channel access is a coalesced b128. Output is never re-read -> NT
// stores; sum reads may still hit L2 from the atomic pass -> default hint.
// ---------------------------------------------------------------------------
__global__ __launch_bounds__(256) void uvnet_normalize(
    float* __restrict__ out,
    const float* __restrict__ wsum,
    const float* __restrict__ sxn,
    const float* __restrict__ syn)
{
    const int g0 = (blockIdx.x * 256 + threadIdx.x) * 4;   // first bin of group
    const int b  = g0 >> 12;
    const int s0 = g0 & 4095;                              // multiple of 4

    const v4f w = *(const v4f*)(wsum + g0);
    v4f inv;
    #pragma unroll
    for (int j = 0; j < 4; ++j) inv[j] = 1.0f / (w[j] + 1e-8f);

    float* o = out + (((size_t)b * OC_) << 12) + s0;
    #pragma unroll 4
    for (int c = 0; c < C_; ++c) {
        v4f v = *(const v4f*)o;
        __builtin_nontemporal_store(v * inv, (v4f*)o);
        o += HW_;
    }
    const v4f xs = *(const v4f*)(sxn + g0);
    const v4f ys = *(const v4f*)(syn + g0);
    __builtin_nontemporal_store(xs * inv, (v4f*)o); o += HW_;   // ch 256: xn
    __builtin_nontemporal_store(ys * inv, (v4f*)o); o += HW_;   // ch 257: yn
    v4f m;
    #pragma unroll
    for (int j = 0; j < 4; ++j) m[j] = w[j] > 0.0f ? 1.0f : 0.0f;
    __builtin_nontemporal_store(m, (v4f*)o);                    // ch 258: mask
}

extern "C" void kernel_launch(void* const* d_in, const int* in_sizes, int n_in,
                              void* d_out, int out_size, void* d_ws, size_t ws_size,
                              hipStream_t stream) {
    const float* dp   = (const float*)d_in[0];   // (64,3,64,64) fp32
    const float* feat = (const float*)d_in[1];   // (64,256,64,64) fp32
    // d_in[2] is uv_res == 64 (baked into the constants above)

    float* out  = (float*)d_out;
    float* wsum = (float*)d_ws;                  // 1 MiB
    float* sxn  = wsum + NBINS;                  // 1 MiB
    float* syn  = sxn  + NBINS;                  // 1 MiB

    // Zero the accumulation targets (graph-capture-safe async memsets).
    hipMemsetAsync(d_out, 0, (size_t)out_size * sizeof(float), stream);
    hipMemsetAsync(d_ws,  0, (size_t)3 * NBINS * sizeof(float), stream);

    dim3 gScatter(NPIX / 4 / 256, C_ / CH_PER_THREAD);   // 256 x 8 blocks
    uvnet_scatter<<<gScatter, 256, 0, stream>>>(dp, feat, out, wsum, sxn, syn);

    uvnet_normalize<<<NBINS / 4 / 256, 256, 0, stream>>>(out, wsum, sxn, syn);
}